// CombinedLoss_29703993819945
// MI455X (gfx1250) — compile-verified
//
#include <hip/hip_runtime.h>
#include <hip/hip_bf16.h>

// ---------------------------------------------------------------------------
// CombinedLoss (NCC + edge-NCC + shape-NCC) for MI455X / gfx1250.
//
// (D,H,W) = (160,192,160) fp32. Bandwidth-bound: after fusing the z-axis box
// sum with the cc evaluation/reduction, total traffic is ~1.3 GB @ 23.3 TB/s
// (~55us floor), working set ~L2-sized. fp32 precision is required by the NCC
// variance cancellation -> V_WMMA_F32_16X16X4_F32 is the CDNA5 matrix op:
// 9-tap box sums along strided axes (y,z) as banded ones-matrix matmuls
//   Out(16 axis-offsets x 16 x's) = Band(16x24) x In(24x16)
// 6 chained wmma per tile, 2 x-tiles per wave; the fused z-pass runs all 5
// statistic volumes per wave (60 wmma/wave) and reduces cc in-register.
//
// Boundary handling: load addresses are CLAMPED (always-valid, unconditional,
// pipelineable loads); zeroing is carried by the banded A matrix via a
// scalar 24-bit valid-window mask ANDed with the per-lane band mask.
// ---------------------------------------------------------------------------

#define DD 160
#define HH 192
#define WW 160
#define VOL (DD * HH * WW)          // 4,915,200
#define NBLK_CC 1200                // (WW/32)*(DD/16)*(HH/8) fused-z blocks

typedef __attribute__((ext_vector_type(2))) float v2f;
typedef __attribute__((ext_vector_type(8))) float v8f;

// ---------------------------------------------------------------- helpers ---
// Branch-free zero-padded fetch: clamped (always valid) address + select.
__device__ __forceinline__ float getv(const float* __restrict__ v,
                                      int z, int y, int x) {
  const bool ok = (unsigned)z < (unsigned)DD && (unsigned)y < (unsigned)HH &&
                  (unsigned)x < (unsigned)WW;
  const int cz = min(max(z, 0), DD - 1);
  const int cy = min(max(y, 0), HH - 1);
  const int cx = min(max(x, 0), WW - 1);
  const float val = v[(cz * HH + cy) * WW + cx];
  return ok ? val : 0.0f;
}

// Reference NCC cc term from the five 9^3 window sums.
__device__ __forceinline__ float cc_term(float S1, float S2, float S11,
                                         float S22, float S12) {
  const float wsz = 729.0f;                    // 9^3
  float uI = S1 / wsz, uJ = S2 / wsz;
  float cross = S12 - uJ * S1 - uI * S2 + uI * uJ * wsz;
  float Ivar  = S11 - 2.0f * uI * S1 + uI * uI * wsz;
  float Jvar  = S22 - 2.0f * uJ * S2 + uJ * uJ * wsz;
  return cross * cross / (Ivar * Jvar + 1e-5f);
}

// -------------------------------------------------------- products kernels ---
// Write 5 volumes into b0: [I, J, I*I, J*J, I*J]
__global__ void products_plain(const float* __restrict__ T,
                               const float* __restrict__ P,
                               float* __restrict__ b0) {
  int i = blockIdx.x * blockDim.x + threadIdx.x;
  if (i >= VOL) return;
  float t = T[i], p = P[i];
  b0[i]           = t;
  b0[VOL + i]     = p;
  b0[2 * VOL + i] = t * t;
  b0[3 * VOL + i] = p * p;
  b0[4 * VOL + i] = t * p;
}

__global__ void products_lap(const float* __restrict__ T,
                             const float* __restrict__ P,
                             float* __restrict__ b0) {
  int i = blockIdx.x * blockDim.x + threadIdx.x;
  if (i >= VOL) return;
  int x = i % WW;
  int r = i / WW;
  int y = r % HH;
  int z = r / HH;
  float lt = 6.0f * getv(T, z, y, x)
           - getv(T, z - 1, y, x) - getv(T, z + 1, y, x)
           - getv(T, z, y - 1, x) - getv(T, z, y + 1, x)
           - getv(T, z, y, x - 1) - getv(T, z, y, x + 1);
  float lp = 6.0f * getv(P, z, y, x)
           - getv(P, z - 1, y, x) - getv(P, z + 1, y, x)
           - getv(P, z, y - 1, x) - getv(P, z, y + 1, x)
           - getv(P, z, y, x - 1) - getv(P, z, y, x + 1);
  b0[i]           = lt;
  b0[VOL + i]     = lp;
  b0[2 * VOL + i] = lt * lt;
  b0[3 * VOL + i] = lp * lp;
  b0[4 * VOL + i] = lt * lp;
}

__global__ void products_shape(const float* __restrict__ T,
                               const float* __restrict__ P,
                               float* __restrict__ b0) {
  int i = blockIdx.x * blockDim.x + threadIdx.x;
  if (i >= VOL) return;
  int x = i % WW;
  int r = i / WW;
  int y = r % HH;
  int z = r / HH;

  // Reference Sobel kernels factor as outer products (signs squared away):
  //   gx: u[kd]*s[kh]*d[kw];  gy: d[kd]*u[kh]*s[kw];  gz: s[kd]*d[kh]*u[kw]
  const float s3[3] = {1.0f, 2.0f, 1.0f};
  const float d3[3] = {-1.0f, 0.0f, 1.0f};

  float gxT = 0.f, gyT = 0.f, gzT = 0.f;
  float gxP = 0.f, gyP = 0.f, gzP = 0.f;
#pragma unroll
  for (int kd = 0; kd < 3; ++kd)
#pragma unroll
    for (int kh = 0; kh < 3; ++kh)
#pragma unroll
      for (int kw = 0; kw < 3; ++kw) {
        float wx = s3[kh] * d3[kw];
        float wy = d3[kd] * s3[kw];
        float wz = s3[kd] * d3[kh];
        if (wx == 0.f && wy == 0.f && wz == 0.f) continue;
        float vt = getv(T, z + kd - 1, y + kh - 1, x + kw - 1);
        float vp = getv(P, z + kd - 1, y + kh - 1, x + kw - 1);
        gxT += wx * vt; gyT += wy * vt; gzT += wz * vt;
        gxP += wx * vp; gyP += wy * vp; gzP += wz * vp;
      }
  float mt = sqrtf(gxT * gxT + gyT * gyT + gzT * gzT);
  float mp = sqrtf(gxP * gxP + gyP * gyP + gzP * gzP);
  b0[i]           = mt;
  b0[VOL + i]     = mp;
  b0[2 * VOL + i] = mt * mt;
  b0[3 * VOL + i] = mp * mp;
  b0[4 * VOL + i] = mt * mp;
}

// ------------------------------------------------- box sum along x (plain) ---
// 9 cached, coalesced, UNCONDITIONAL taps (clamped address + mask-fma).
__global__ void boxsum_x(const float* __restrict__ src,
                         float* __restrict__ dst) {
  int i = blockIdx.x * blockDim.x + threadIdx.x;   // grid covers 5*VOL exactly
  int x = i % WW;
  int rowbase = i - x;
  float s = 0.0f;
#pragma unroll
  for (int j = -4; j <= 4; ++j) {
    int xx = x + j;
    int cx = min(max(xx, 0), WW - 1);
    float m = ((unsigned)xx < (unsigned)WW) ? 1.0f : 0.0f;
    s = fmaf(m, src[rowbase + cx], s);
  }
  dst[i] = s;
}

// --------------------------- y-axis box sum (WMMA banded matmul, fp32) -------
// One wave: TWO 16(axis-offset M) x 16(x N) tiles sharing banded-A fragments.
__global__ void boxsum_y_wmma(const float* __restrict__ src,
                              float* __restrict__ dst) {
  const int lane = threadIdx.x & 31;
  const int wave = threadIdx.x >> 5;           // 8 waves per block
  const int x0   = blockIdx.x * 32;
  const int a0   = blockIdx.y * 16;            // y tile
  const int vol  = blockIdx.z / (DD / 8);      // which of the 5 volumes
  const int oc   = blockIdx.z % (DD / 8);
  const int z    = oc * 8 + wave;
  const int base = vol * VOL + z * (WW * HH);

  const int n    = lane & 15;
  const int half = lane >> 4;

  // Uniform valid-t window (t in [0,23], row a0-4+t must be in [0,HH)) as a
  // scalar bitmask; per-lane band rows n..n+8 AND it.  A element = one bit.
  const int lo = max(0, 4 - a0);
  const int hi = min(23, HH + 3 - a0);
  const unsigned validMask = ((1u << (hi + 1)) - 1) & ~((1u << lo) - 1);
  const unsigned laneMask  = (0x1FFu << n) & validMask;

  v2f aF[6];
#pragma unroll
  for (int kk = 0; kk < 6; ++kk) {
    const int kA0 = kk * 4 + 2 * half;         // A VGPR0: K = 4kk + {0|2}
    aF[kk].x = ((laneMask >> kA0) & 1u) ? 1.0f : 0.0f;
    aF[kk].y = ((laneMask >> (kA0 + 1)) & 1u) ? 1.0f : 0.0f;
  }

  // Preload all B fragments (24 unconditional coalesced loads, clamped rows).
  float bL[12], bR[12];
#pragma unroll
  for (int kk = 0; kk < 6; ++kk) {
    const int aa0 = a0 - 4 + kk * 4 + half;    // B VGPR0: K = 4kk + {0|1}
    const int aa1 = aa0 + 2;                   // B VGPR1: K = 4kk + {2|3}
    const int p0  = base + min(max(aa0, 0), HH - 1) * WW + x0 + n;
    const int p1  = base + min(max(aa1, 0), HH - 1) * WW + x0 + n;
    bL[2 * kk]     = src[p0];
    bL[2 * kk + 1] = src[p1];
    bR[2 * kk]     = src[p0 + 16];
    bR[2 * kk + 1] = src[p1 + 16];
  }

  v8f c0 = {0.f, 0.f, 0.f, 0.f, 0.f, 0.f, 0.f, 0.f};
  v8f c1 = {0.f, 0.f, 0.f, 0.f, 0.f, 0.f, 0.f, 0.f};
#pragma unroll
  for (int kk = 0; kk < 6; ++kk) {
    v2f b;
    b.x = bL[2 * kk]; b.y = bL[2 * kk + 1];
    c0 = __builtin_amdgcn_wmma_f32_16x16x4_f32(
        false, aF[kk], false, b, (short)0, c0, false, false);
    b.x = bR[2 * kk]; b.y = bR[2 * kk + 1];
    c1 = __builtin_amdgcn_wmma_f32_16x16x4_f32(
        false, aF[kk], false, b, (short)0, c1, false, false);
  }

#pragma unroll
  for (int r2 = 0; r2 < 8; ++r2) {
    const int mo = r2 + 8 * half;
    const int o  = base + (a0 + mo) * WW + x0 + n;
    dst[o]      = c0[r2];
    dst[o + 16] = c1[r2];
  }
}

// ------------- FUSED z-axis box sum + cc + block reduction (WMMA, fp32) ------
// One wave runs the banded z-matmul for ALL FIVE statistic volumes
// (5 x 12 = 60 wmma) so S1,S2,S11,S22,S12 for its 16 voxels stay in
// registers; cc is evaluated in-register and block-reduced to partials.
// Eliminates one full 5-volume write + 5-volume read (~590 MB over 3 passes).
__global__ void boxsum_z_cc_wmma(const float* __restrict__ src,
                                 float* __restrict__ partials) {
  const int lane = threadIdx.x & 31;
  const int wave = threadIdx.x >> 5;           // 8 waves per block
  const int x0   = blockIdx.x * 32;
  const int a0   = blockIdx.y * 16;            // z tile
  const int y    = blockIdx.z * 8 + wave;
  const int base = y * WW;

  const int n    = lane & 15;
  const int half = lane >> 4;

  const int lo = max(0, 4 - a0);
  const int hi = min(23, DD + 3 - a0);
  const unsigned validMask = ((1u << (hi + 1)) - 1) & ~((1u << lo) - 1);
  const unsigned laneMask  = (0x1FFu << n) & validMask;

  v2f aF[6];
#pragma unroll
  for (int kk = 0; kk < 6; ++kk) {
    const int kA0 = kk * 4 + 2 * half;
    aF[kk].x = ((laneMask >> kA0) & 1u) ? 1.0f : 0.0f;
    aF[kk].y = ((laneMask >> (kA0 + 1)) & 1u) ? 1.0f : 0.0f;
  }

  // Clamped row offsets, shared by all 5 volumes.
  int o0[6], o1[6];
#pragma unroll
  for (int kk = 0; kk < 6; ++kk) {
    const int aa0 = a0 - 4 + kk * 4 + half;
    const int aa1 = aa0 + 2;
    o0[kk] = base + min(max(aa0, 0), DD - 1) * (WW * HH) + x0 + n;
    o1[kk] = base + min(max(aa1, 0), DD - 1) * (WW * HH) + x0 + n;
  }

  v8f cL[5], cR[5];
#pragma unroll
  for (int v = 0; v < 5; ++v) {
    const float* __restrict__ s = src + v * VOL;
    float bL[12], bR[12];
#pragma unroll
    for (int kk = 0; kk < 6; ++kk) {
      bL[2 * kk]     = s[o0[kk]];
      bL[2 * kk + 1] = s[o1[kk]];
      bR[2 * kk]     = s[o0[kk] + 16];
      bR[2 * kk + 1] = s[o1[kk] + 16];
    }
    v8f c0 = {0.f, 0.f, 0.f, 0.f, 0.f, 0.f, 0.f, 0.f};
    v8f c1 = {0.f, 0.f, 0.f, 0.f, 0.f, 0.f, 0.f, 0.f};
#pragma unroll
    for (int kk = 0; kk < 6; ++kk) {
      v2f b;
      b.x = bL[2 * kk]; b.y = bL[2 * kk + 1];
      c0 = __builtin_amdgcn_wmma_f32_16x16x4_f32(
          false, aF[kk], false, b, (short)0, c0, false, false);
      b.x = bR[2 * kk]; b.y = bR[2 * kk + 1];
      c1 = __builtin_amdgcn_wmma_f32_16x16x4_f32(
          false, aF[kk], false, b, (short)0, c1, false, false);
    }
    cL[v] = c0;
    cR[v] = c1;
  }

  // cc over this lane's 16 voxels, then block reduction.
  float ccsum = 0.0f;
#pragma unroll
  for (int r2 = 0; r2 < 8; ++r2) {
    ccsum += cc_term(cL[0][r2], cL[1][r2], cL[2][r2], cL[3][r2], cL[4][r2]);
    ccsum += cc_term(cR[0][r2], cR[1][r2], cR[2][r2], cR[3][r2], cR[4][r2]);
  }

  __shared__ float sm[256];
  sm[threadIdx.x] = ccsum;
  __syncthreads();
#pragma unroll
  for (int s = 128; s > 0; s >>= 1) {
    if (threadIdx.x < s) sm[threadIdx.x] += sm[threadIdx.x + s];
    __syncthreads();
  }
  if (threadIdx.x == 0) {
    const int bid = blockIdx.x +
                    gridDim.x * (blockIdx.y + gridDim.y * blockIdx.z);
    partials[bid] = sm[0];
  }
}

// Deterministic final reduction: fixed strided order, single block.
__global__ void finalize(const float* __restrict__ partials,
                         float* __restrict__ out) {
  __shared__ float sm[256];
  float loss[3];
#pragma unroll
  for (int p = 0; p < 3; ++p) {
    float acc = 0.0f;
    for (int j = threadIdx.x; j < NBLK_CC; j += 256)
      acc += partials[p * NBLK_CC + j];
    sm[threadIdx.x] = acc;
    __syncthreads();
    for (int s = 128; s > 0; s >>= 1) {
      if (threadIdx.x < s) sm[threadIdx.x] += sm[threadIdx.x + s];
      __syncthreads();
    }
    loss[p] = -sm[0] / (float)VOL;             // -mean(cc)
    __syncthreads();
  }
  if (threadIdx.x == 0)
    out[0] = 0.8f * loss[0] + 0.1f * loss[1] + 0.1f * loss[2];
}

// ------------------------------------------------------------------ launch ---
extern "C" void kernel_launch(void* const* d_in, const int* in_sizes, int n_in,
                              void* d_out, int out_size, void* d_ws,
                              size_t ws_size, hipStream_t stream) {
  (void)in_sizes; (void)n_in; (void)out_size; (void)ws_size;
  const float* T = (const float*)d_in[0];      // y_true
  const float* P = (const float*)d_in[1];      // y_pred
  float* ws = (float*)d_ws;
  float* buf0 = ws;                            // 5 volumes
  float* buf1 = ws + 5 * (size_t)VOL;          // 5 volumes (ping-pong)
  float* partials = ws + 10 * (size_t)VOL;     // 3 * NBLK_CC floats

  const dim3 blk(256);
  const dim3 gProd(VOL / 256);                 // 19200
  const dim3 gX(5 * VOL / 256);                // 96000
  // y-axis: x tiles 160/32=5; axis tiles 192/16=12; other=z (20 oc) x 5 vols
  const dim3 gY(WW / 32, HH / 16, (DD / 8) * 5);
  // fused z-axis: x tiles 5; axis tiles 160/16=10; other=y (192/8=24) -> 1200
  const dim3 gZ(WW / 32, DD / 16, HH / 8);

  for (int p = 0; p < 3; ++p) {
    if (p == 0)      products_plain<<<gProd, blk, 0, stream>>>(T, P, buf0);
    else if (p == 1) products_lap  <<<gProd, blk, 0, stream>>>(T, P, buf0);
    else             products_shape<<<gProd, blk, 0, stream>>>(T, P, buf0);

    boxsum_x<<<gX, blk, 0, stream>>>(buf0, buf1);
    boxsum_y_wmma<<<gY, blk, 0, stream>>>(buf1, buf0);
    boxsum_z_cc_wmma<<<gZ, blk, 0, stream>>>(buf0, partials + p * NBLK_CC);
  }
  finalize<<<1, blk, 0, stream>>>(partials, (float*)d_out);
}